// Network_36232344109329
// MI455X (gfx1250) — compile-verified
//
#include <hip/hip_runtime.h>
#include <initializer_list>

// ---------------------------------------------------------------------------
// GraphNet forward on MI455X (gfx1250), fp32 via V_WMMA_F32_16X16X4_F32.
// Both GEMM operands staged in LDS; B fragments pre-packed so the inner loop
// is b64 DS loads + WMMA only (no VALU repacking).
// ---------------------------------------------------------------------------

#define N_NODES 20000
#define N_EDGES 160000
#define NGRAPH  16
#define STEPS   2
#define LN_EPS  1e-5f

typedef __attribute__((ext_vector_type(2))) float v2f;
typedef __attribute__((ext_vector_type(8))) float v8f;

struct Seg {
  const float* ptr;   // source matrix [*, width]
  const int*   idx;   // gather indices (nullptr = identity)
  int width;          // feature width of this segment
  int off;            // column offset in concatenated input (filled by host)
};

struct BlockArgs {
  int nrows, din, nseg;
  Seg segs[8];
  const float* W;      // [din, DOUT]
  const float* bias;   // [DOUT]
  const float* gamma;  // [DOUT]
  const float* beta;   // [DOUT]
  float* y_out;        // [nrows, DOUT] (STORE_Y)
  const float* wout;   // [DOUT] (PROJ)
  const float* bout;   // [1]    (PROJ)
  float* proj_out;     // [nrows] (PROJ)
};

#define KC 64
#define KPITCH 68   // input tile pitch: 272B rows -> 16B-aligned float4, A reads conflict-free

__device__ inline float hsum16(float s) {
  // reduce across the 16-lane half-wave (masks 1,2,4,8 stay inside the half)
  s += __shfl_xor(s, 1);
  s += __shfl_xor(s, 2);
  s += __shfl_xor(s, 4);
  s += __shfl_xor(s, 8);
  return s;
}

// Fused: y = LN(ReLU(concat(segs) @ W + b)); optionally proj_out = y @ wout + bout.
// Block = 128 threads (4 waves); each wave computes a 16 x DOUT tile via
// v_wmma_f32_16x16x4_f32, K stepped by 4 over LDS-staged input+weight chunks.
template <int NT, bool STORE_Y, bool PROJ>
__global__ __launch_bounds__(128) void block_kernel(BlockArgs a) {
  constexpr int DOUT = NT * 16;
  // B fragments packed per K-pair row: sWp[k2*PP + 2n + j] = W[2*k2 + j][n].
  // PP % 64 == 32 -> the two half-waves read disjoint 32-bank windows.
  constexpr int PP = 2 * DOUT + 32;
  __shared__ float sIn[64 * KPITCH];
  __shared__ float sWp[(KC / 2) * PP];

  const int tid  = threadIdx.x;
  const int lane = tid & 31;
  const int wave = tid >> 5;
  const int half = lane >> 4;   // which 16-lane half
  const int ln   = lane & 15;
  const int mblock = blockIdx.x * 64;

  v8f acc[NT];
  for (int t = 0; t < NT; ++t)
    for (int r = 0; r < 8; ++r) acc[t][r] = 0.f;

  for (int kc = 0; kc < a.din; kc += KC) {
    const int kw = min(KC, a.din - kc);

    // ---- stage assembled input chunk [64 rows x kw] into LDS (float4) ----
#pragma unroll
    for (int s = 0; s < 8; ++s) {
      if (s >= a.nseg) break;
      const Seg sg = a.segs[s];
      const int c0 = max(sg.off, kc);
      const int c1 = min(sg.off + sg.width, kc + kw);
      const int cw = c1 - c0;
      if (cw <= 0) continue;
      // overlaps are always pow2 multiples of 8, 32-aligned within the chunk
      const int cw4 = cw >> 2;
      const int sh4 = __builtin_ctz(cw4);
      const int total4 = cw4 << 6;           // 64 rows * cw4 quads
      const int lbase = (c0 - kc);
      for (int l = tid; l < total4; l += 128) {
        const int r  = l >> sh4;
        const int c4 = (l & (cw4 - 1)) << 2;
        const int grow = mblock + r;
        float4 v = make_float4(0.f, 0.f, 0.f, 0.f);
        if (grow < a.nrows) {
          const int srow = sg.idx ? sg.idx[grow] : grow;
          v = *(const float4*)(sg.ptr + (long)srow * sg.width + (c0 - sg.off) + c4);
        }
        *(float4*)(&sIn[r * KPITCH + lbase + c4]) = v;
      }
    }
    // ---- stage weight chunk [kw x DOUT] into LDS, packed per K-pair ----
    {
      const int total2 = (kw >> 1) * DOUT;   // one element = K-pair for one n
      const float* Wc = a.W + (long)kc * DOUT;
      for (int l = tid; l < total2; l += 128) {
        const int k2 = l >> (NT == 8 ? 7 : 5);   // l / DOUT
        const int n  = l & (DOUT - 1);
        v2f v;
        v.x = Wc[(long)(2 * k2) * DOUT + n];
        v.y = Wc[(long)(2 * k2 + 1) * DOUT + n];
        *(v2f*)(&sWp[k2 * PP + 2 * n]) = v;
      }
    }
    __syncthreads();

    // ---- WMMA over this K chunk: b64 DS loads + WMMA only ----
    const int arow = (16 * wave + ln) * KPITCH;
    for (int k4 = 0; k4 < kw; k4 += 4) {
      // A: (M=ln, K = k4 + 2*half + j) -> one 8B LDS load
      const v2f av = *(const v2f*)(&sIn[arow + k4 + 2 * half]);
      // B: (K = k4 + 2*half + j, N = 16t + ln) -> packed, one 8B load per tile
      const float* Wl = &sWp[((k4 >> 1) + half) * PP + 2 * ln];
#pragma unroll
      for (int t = 0; t < NT; ++t) {
        const v2f bv = *(const v2f*)(Wl + 32 * t);
        acc[t] = __builtin_amdgcn_wmma_f32_16x16x4_f32(
            false, av, false, bv, (short)0, acc[t], false, false);
      }
    }
    __syncthreads();
  }

  // ---- bias + ReLU ----
  for (int t = 0; t < NT; ++t) {
    const float b = a.bias[t * 16 + ln];
    for (int r = 0; r < 8; ++r) {
      const float h = acc[t][r] + b;
      acc[t][r] = h > 0.f ? h : 0.f;
    }
  }

  // ---- LayerNorm across the row (row M = r + 8*half lives in one half-wave) ----
  float mu[8], rstd[8];
  for (int r = 0; r < 8; ++r) {
    float s = 0.f;
    for (int t = 0; t < NT; ++t) s += acc[t][r];
    mu[r] = hsum16(s) / (float)DOUT;
  }
  for (int r = 0; r < 8; ++r) {
    float s = 0.f;
    for (int t = 0; t < NT; ++t) {
      const float d = acc[t][r] - mu[r];
      s += d * d;
    }
    rstd[r] = rsqrtf(hsum16(s) / (float)DOUT + LN_EPS);
  }
  for (int t = 0; t < NT; ++t) {
    const float g  = a.gamma[t * 16 + ln];
    const float be = a.beta[t * 16 + ln];
    for (int r = 0; r < 8; ++r)
      acc[t][r] = g * (acc[t][r] - mu[r]) * rstd[r] + be;
  }

  if (STORE_Y) {
    for (int r = 0; r < 8; ++r) {
      const int grow = mblock + 16 * wave + 8 * half + r;
      if (grow < a.nrows)
        for (int t = 0; t < NT; ++t)
          a.y_out[(long)grow * DOUT + t * 16 + ln] = acc[t][r];
    }
  }
  if (PROJ) {
    float p[8];
    for (int r = 0; r < 8; ++r) {
      float s = 0.f;
      for (int t = 0; t < NT; ++t) s += acc[t][r] * a.wout[t * 16 + ln];
      p[r] = hsum16(s);
    }
    if (ln == 0) {
      const float bo = a.bout[0];
      for (int r = 0; r < 8; ++r) {
        const int grow = mblock + 16 * wave + 8 * half + r;
        if (grow < a.nrows) a.proj_out[grow] = p[r] + bo;
      }
    }
  }
}

// agg[dst[i], :] += e[i, :]   (atomics over 2.56M addresses -> low contention)
__global__ void scatter_add_edges(const float* __restrict__ e,
                                  const int* __restrict__ dst,
                                  float* __restrict__ agg, int nE) {
  const long i = (long)blockIdx.x * blockDim.x + threadIdx.x;
  if (i >= (long)nE * 128) return;
  const int row = (int)(i >> 7);
  const int c   = (int)(i & 127);
  atomicAdd(agg + ((long)dst[row] << 7) + c, e[i]);
}

// segment-sum of [nrows,128] into [16,128]; idx is SORTED -> register
// accumulation with ~1 atomic flush per (thread, segment boundary).
__global__ void segsum16_kernel(const float* __restrict__ data,
                                const int* __restrict__ idx,
                                float* __restrict__ out, int nrows) {
  const int streams = gridDim.x * 2;
  const int sid = blockIdx.x * 2 + (threadIdx.x >> 7);
  const int c   = threadIdx.x & 127;
  const int per = (nrows + streams - 1) / streams;
  const int r0 = sid * per;
  const int r1 = min(nrows, r0 + per);
  if (r0 >= r1) return;
  float acc = 0.f;
  int cur = idx[r0];
  for (int r = r0; r < r1; ++r) {
    const int s = idx[r];
    if (s != cur) {
      atomicAdd(out + cur * 128 + c, acc);
      acc = 0.f;
      cur = s;
    }
    acc += data[(long)r * 128 + c];
  }
  atomicAdd(out + cur * 128 + c, acc);
}

// ---------------------------------------------------------------------------
// Host side
// ---------------------------------------------------------------------------

struct BlkP { const float *W, *b, *g, *beta; };
struct LinP { const float *W, *b; };

static BlockArgs make_args(int nrows, int din, const BlkP& p, float* y_out,
                           const float* wout, const float* bout, float* proj_out,
                           std::initializer_list<Seg> segs) {
  BlockArgs a = {};
  a.nrows = nrows;
  a.din = din;
  a.nseg = (int)segs.size();
  int i = 0, off = 0;
  for (Seg s : segs) {
    s.off = off;
    off += s.width;
    a.segs[i++] = s;
  }
  a.W = p.W; a.bias = p.b; a.gamma = p.g; a.beta = p.beta;
  a.y_out = y_out; a.wout = wout; a.bout = bout; a.proj_out = proj_out;
  return a;
}

static void run_block(hipStream_t st, int NT, bool store_y, const BlockArgs& a) {
  const int grid = (a.nrows + 63) / 64;
  if (NT == 8) {
    if (store_y) block_kernel<8, true,  false><<<grid, 128, 0, st>>>(a);
    else         block_kernel<8, false, true ><<<grid, 128, 0, st>>>(a);
  } else {
    if (store_y) block_kernel<2, true,  false><<<grid, 128, 0, st>>>(a);
    else         block_kernel<2, false, true ><<<grid, 128, 0, st>>>(a);
  }
}

extern "C" void kernel_launch(void* const* d_in, const int* in_sizes, int n_in,
                              void* d_out, int out_size, void* d_ws, size_t ws_size,
                              hipStream_t stream) {
  (void)out_size; (void)ws_size;
  auto F = [&](int i) { return (const float*)d_in[i]; };
  auto I = [&](int i) { return (const int*)d_in[i]; };

  const float *x_raw, *e_raw, *g_raw;
  const int *edges, *node_idx, *edge_idx;
  BlkP enc_e, enc_x, enc_g, core_e, core_x, core_g;
  BlkP dec_e0, dec_e1, dec_x0, dec_x1, dec_g0;
  LinP out_e, out_x, out_g;

  if (n_in == 8) {
    // params concatenated flat at slot 3, leaves in jax-sorted order
    x_raw = F(0); e_raw = F(1); g_raw = F(2);
    const float* P = F(3);
    size_t o = 0;
    auto take = [&](size_t n) { const float* p = P + o; o += n; return p; };
    auto blkT = [&](int di, int dd) {
      BlkP r; r.W = take((size_t)di * dd); r.b = take(dd); r.beta = take(dd); r.g = take(dd); return r;
    };
    core_e = blkT(832, 128); core_g = blkT(320, 32); core_x = blkT(448, 128);
    dec_e0 = blkT(128, 128); dec_e1 = blkT(128, 128); dec_g0 = blkT(32, 32);
    dec_x0 = blkT(128, 128); dec_x1 = blkT(128, 128);
    enc_e = blkT(32, 128); enc_g = blkT(8, 32); enc_x = blkT(64, 128);
    out_e.W = take(128); out_e.b = take(1);
    out_g.W = take(32);  out_g.b = take(1);
    out_x.W = take(128); out_x.b = take(1);
    edges = I(4); node_idx = I(5); edge_idx = I(6);
  } else if (in_sizes[0] == 5120000) {
    // full jax pytree flatten: dict keys sorted at every level
    e_raw = F(0); edge_idx = I(1); edges = I(2); g_raw = F(3); node_idx = I(4);
    auto blkS = [&](int b) { return BlkP{F(b), F(b + 1), F(b + 3), F(b + 2)}; };  // W,b,beta,g
    core_e = blkS(5); core_g = blkS(9); core_x = blkS(13);
    dec_e0 = blkS(17); dec_e1 = blkS(21); dec_g0 = blkS(25);
    dec_x0 = blkS(29); dec_x1 = blkS(33);
    enc_e = blkS(37); enc_g = blkS(41); enc_x = blkS(45);
    out_e = {F(49), F(50)}; out_g = {F(51), F(52)}; out_x = {F(53), F(54)};
    x_raw = F(56);
  } else {
    // insertion-order flatten (dict insertion order, blocks as W,b,g,beta)
    x_raw = F(0); e_raw = F(1); g_raw = F(2);
    auto blkI = [&](int b) { return BlkP{F(b), F(b + 1), F(b + 2), F(b + 3)}; };
    enc_e = blkI(3); enc_x = blkI(7); enc_g = blkI(11);
    core_e = blkI(15); core_x = blkI(19); core_g = blkI(23);
    dec_e0 = blkI(27); dec_e1 = blkI(31);
    dec_x0 = blkI(35); dec_x1 = blkI(39);
    dec_g0 = blkI(43);
    out_e = {F(47), F(48)}; out_x = {F(49), F(50)}; out_g = {F(51), F(52)};
    edges = I(53); node_idx = I(54); edge_idx = I(55);
  }

  // ---- workspace layout (fp32), ~287 MB ----
  float* ws = (float*)d_ws;
  size_t off = 0;
  auto alloc = [&](size_t n) { float* p = ws + off; off += n; return p; };
  float* e0   = alloc((size_t)N_EDGES * 128);
  float* eA   = alloc((size_t)N_EDGES * 128);
  float* eB   = alloc((size_t)N_EDGES * 128);
  float* x0   = alloc((size_t)N_NODES * 128);
  float* xA   = alloc((size_t)N_NODES * 128);
  float* xB   = alloc((size_t)N_NODES * 128);
  float* aggE = alloc((size_t)N_NODES * 128);
  float* g0   = alloc((size_t)NGRAPH * 32);
  float* gA   = alloc((size_t)NGRAPH * 32);
  float* gB   = alloc((size_t)NGRAPH * 32);
  float* nagg = alloc((size_t)NGRAPH * 128);
  float* eagg = alloc((size_t)NGRAPH * 128);

  float* out    = (float*)d_out;
  float* out_xs = out;                                      // [2, N_NODES]
  float* out_es = out + (size_t)STEPS * N_NODES;            // [2, N_EDGES]
  float* out_gs = out + (size_t)STEPS * (N_NODES + N_EDGES);// [2, 16]

  const int* src = edges;
  const int* dst = edges + N_EDGES;

  // ---- encoders ----
  run_block(stream, 8, true, make_args(N_EDGES, 32, enc_e, e0, nullptr, nullptr, nullptr,
            {{e_raw, nullptr, 32, 0}}));
  run_block(stream, 8, true, make_args(N_NODES, 64, enc_x, x0, nullptr, nullptr, nullptr,
            {{x_raw, nullptr, 64, 0}}));
  run_block(stream, 2, true, make_args(NGRAPH, 8, enc_g, g0, nullptr, nullptr, nullptr,
            {{g_raw, nullptr, 8, 0}}));
  hipMemcpyAsync(eA, e0, (size_t)N_EDGES * 128 * 4, hipMemcpyDeviceToDevice, stream);
  hipMemcpyAsync(xA, x0, (size_t)N_NODES * 128 * 4, hipMemcpyDeviceToDevice, stream);
  hipMemcpyAsync(gA, g0, (size_t)NGRAPH * 32 * 4, hipMemcpyDeviceToDevice, stream);

  float *eCur = eA, *eNew = eB, *xCur = xA, *xNew = xB, *gCur = gA, *gNew = gB;

  for (int s = 0; s < STEPS; ++s) {
    // core_e: cat(e0,e, x0[src],x[src], x0[dst],x[dst], g0[ei],g[ei]) -> 832
    run_block(stream, 8, true, make_args(N_EDGES, 832, core_e, eNew, nullptr, nullptr, nullptr,
              {{e0, nullptr, 128, 0}, {eCur, nullptr, 128, 0},
               {x0, src, 128, 0},     {xCur, src, 128, 0},
               {x0, dst, 128, 0},     {xCur, dst, 128, 0},
               {g0, edge_idx, 32, 0}, {gCur, edge_idx, 32, 0}}));

    // agg_e = segment_sum(eNew, dst, N_NODES)
    hipMemsetAsync(aggE, 0, (size_t)N_NODES * 128 * 4, stream);
    {
      const long tot = (long)N_EDGES * 128;
      scatter_add_edges<<<(int)((tot + 255) / 256), 256, 0, stream>>>(eNew, dst, aggE, N_EDGES);
    }

    // core_x: cat(x0,x, agg_e, g0[ni],g[ni]) -> 448
    run_block(stream, 8, true, make_args(N_NODES, 448, core_x, xNew, nullptr, nullptr, nullptr,
              {{x0, nullptr, 128, 0}, {xCur, nullptr, 128, 0}, {aggE, nullptr, 128, 0},
               {g0, node_idx, 32, 0}, {gCur, node_idx, 32, 0}}));

    // graph-level aggregations (sorted indices)
    hipMemsetAsync(nagg, 0, (size_t)NGRAPH * 128 * 4, stream);
    hipMemsetAsync(eagg, 0, (size_t)NGRAPH * 128 * 4, stream);
    segsum16_kernel<<<64, 256, 0, stream>>>(xNew, node_idx, nagg, N_NODES);
    segsum16_kernel<<<256, 256, 0, stream>>>(eNew, edge_idx, eagg, N_EDGES);

    // core_g: cat(g0,g, node_agg, edge_agg) -> 320
    run_block(stream, 2, true, make_args(NGRAPH, 320, core_g, gNew, nullptr, nullptr, nullptr,
              {{g0, nullptr, 32, 0}, {gCur, nullptr, 32, 0},
               {nagg, nullptr, 128, 0}, {eagg, nullptr, 128, 0}}));

    // decoders (t1 reuses the retired ping-pong buffer; blk2 fuses 128->1 proj)
    run_block(stream, 8, true, make_args(N_EDGES, 128, dec_e0, eCur, nullptr, nullptr, nullptr,
              {{eNew, nullptr, 128, 0}}));
    run_block(stream, 8, false, make_args(N_EDGES, 128, dec_e1, nullptr, out_e.W, out_e.b,
              out_es + (size_t)s * N_EDGES, {{eCur, nullptr, 128, 0}}));

    run_block(stream, 8, true, make_args(N_NODES, 128, dec_x0, xCur, nullptr, nullptr, nullptr,
              {{xNew, nullptr, 128, 0}}));
    run_block(stream, 8, false, make_args(N_NODES, 128, dec_x1, nullptr, out_x.W, out_x.b,
              out_xs + (size_t)s * N_NODES, {{xCur, nullptr, 128, 0}}));

    run_block(stream, 2, false, make_args(NGRAPH, 32, dec_g0, nullptr, out_g.W, out_g.b,
              out_gs + (size_t)s * NGRAPH, {{gNew, nullptr, 32, 0}}));

    float* t;
    t = eCur; eCur = eNew; eNew = t;
    t = xCur; xCur = xNew; xNew = t;
    t = gCur; gCur = gNew; gNew = t;
  }
}